// SimpleNet_67662914781854
// MI455X (gfx1250) — compile-verified
//
#include <hip/hip_runtime.h>
#include <math.h>

// ---------------------------------------------------------------------------
// Quanvolutional net for MI455X (gfx1250, wave32).
// Dominant stage: 261121 x (4-qubit sim) rewritten as a fixed 32x32 real
// matrix (precomputed from q_weights) times per-patch 32-vector, mapped onto
// v_wmma_f32_16x16x4_f32 (f32 WMMA keeps reference complex64 precision).
// B-operand values are built purely from scalar trig factors (no private
// register arrays) so the compiler cannot introduce variable-index extract
// select-trees.
// ---------------------------------------------------------------------------

typedef __attribute__((ext_vector_type(2))) float v2f;
typedef __attribute__((ext_vector_type(8))) float v8f;

#define HP 511           // patch grid is 511 x 511
#define WTILES 32        // ceil(511/16) patch tiles per row

// ---------------------------------------------------------------------------
// Kernel 0: build M = [[Re U, -Im U],[Im U, Re U]] (32x32, row-major) where U
// is the fixed 16x16 unitary of the Rot+CNOT layers. Lane j pushes basis
// state e_j through the circuit; its final state is column j of U.
// Args to trig are <= 4*pi: hardware __sincosf is ~1e-6 absolute here.
// ---------------------------------------------------------------------------
__global__ __launch_bounds__(32) void k_build_M(const float* __restrict__ qw,
                                                float* __restrict__ Mg) {
  int j = threadIdx.x;
  if (j >= 16) return;
  float re[16], im[16];
#pragma unroll
  for (int b = 0; b < 16; ++b) { re[b] = (b == j) ? 1.f : 0.f; im[b] = 0.f; }

#pragma unroll
  for (int l = 0; l < 3; ++l) {
    // --- Rot(phi, theta, omega) on each qubit ---
#pragma unroll
    for (int w = 0; w < 4; ++w) {
      float phi = qw[(l * 4 + w) * 3 + 0];
      float th  = qw[(l * 4 + w) * 3 + 1];
      float om  = qw[(l * 4 + w) * 3 + 2];
      float ct, st, ca, sa, cb, sb;
      __sincosf(0.5f * th, &st, &ct);
      __sincosf(0.5f * (phi + om), &sa, &ca);
      __sincosf(0.5f * (phi - om), &sb, &cb);
      // u00 = e^{-i a} c ; u01 = -e^{+i b} s ; u10 = e^{-i b} s ; u11 = e^{+i a} c
      float u00r =  ca * ct, u00i = -sa * ct;
      float u01r = -cb * st, u01i = -sb * st;
      float u10r =  cb * st, u10i = -sb * st;
      float u11r =  ca * ct, u11i =  sa * ct;
      int mask = 8 >> w;  // qubit w is bit (3-w)
#pragma unroll
      for (int b = 0; b < 16; ++b) {
        if (b & mask) continue;
        int b1 = b | mask;
        float a0r = re[b], a0i = im[b], a1r = re[b1], a1i = im[b1];
        re[b]  = u00r * a0r - u00i * a0i + u01r * a1r - u01i * a1i;
        im[b]  = u00r * a0i + u00i * a0r + u01r * a1i + u01i * a1r;
        re[b1] = u10r * a0r - u10i * a0i + u11r * a1r - u11i * a1i;
        im[b1] = u10r * a0i + u10i * a0r + u11r * a1i + u11i * a1r;
      }
    }
    // --- CNOT ring: control w, target (w + r) % 4, r = l % 3 + 1 ---
    int r = l % 3 + 1;
#pragma unroll
    for (int w = 0; w < 4; ++w) {
      int cm = 8 >> w, tm = 8 >> ((w + r) & 3);
#pragma unroll
      for (int b = 0; b < 16; ++b) {
        if ((b & cm) && !(b & tm)) {
          int b2 = b | tm;
          float tr = re[b]; re[b] = re[b2]; re[b2] = tr;
          float ti = im[b]; im[b] = im[b2]; im[b2] = ti;
        }
      }
    }
  }
  // Column j of U -> columns j and j+16 of M.
#pragma unroll
  for (int rr = 0; rr < 16; ++rr) {
    Mg[rr * 32 + j]             =  re[rr];
    Mg[rr * 32 + 16 + j]        = -im[rr];
    Mg[(rr + 16) * 32 + j]      =  im[rr];
    Mg[(rr + 16) * 32 + 16 + j] =  re[rr];
  }
}

// ---------------------------------------------------------------------------
// Kernel 1: qconv. One wave32 = 16 patches. out32 = M * x32 via 16x
// v_wmma_f32_16x16x4_f32, then probs -> 4 expectation values. Stores raw ez
// (sigmoid commutes with the later maxpool).
//
// B element for (kk, h) on lane-half `hi` corresponds to basis index b with
// bits {kk[1], kk[0], hi, h} and Re (kk<4) / Im (kk>=4) part of
// psi_in[b] = (-i)^popcount(b) * prod_w (bit ? sin : cos). The hi dependence
// factors entirely through 4 scalar terms FA..FD, so no register arrays.
// ---------------------------------------------------------------------------
__global__ __launch_bounds__(256) void k_qconv(const float* __restrict__ img,
                                               const float* __restrict__ Mg,
                                               float* __restrict__ ez) {
  int wave = (blockIdx.x * blockDim.x + threadIdx.x) >> 5;
  int lane = threadIdx.x & 31;
  int i  = wave / WTILES;          // patch row
  int jt = wave % WTILES;          // patch-column tile
  if (i >= HP) return;             // wave-uniform (never divergent)
  int n  = lane & 15;              // patch slot within tile / WMMA column
  int j  = jt * 16 + n;
  int jc = j > (HP - 1) ? (HP - 1) : j;   // clamp loads, predicate stores

  // 2x2 patch inputs (a, b, c, d ordering of the reference).
  float x0 = img[i * 512 + jc];
  float x1 = img[i * 512 + jc + 1];
  float x2 = img[(i + 1) * 512 + jc];
  float x3 = img[(i + 1) * 512 + jc + 1];
  float c0, s0, c1, s1, c2, s2, c3, s3;
  __sincosf(0.5f * x0, &s0, &c0);
  __sincosf(0.5f * x1, &s1, &c1);
  __sincosf(0.5f * x2, &s2, &c2);
  __sincosf(0.5f * x3, &s3, &c3);

  bool hi = lane >= 16;
  // Scalar phase/qubit-2 factors; select operands are plain scalars.
  float FA = hi ? 0.f : c2;    // phase q=0 (Re) / q=3 (Im)
  float FB = hi ? -s2 : 0.f;   // phase q=1 (Re) / q=0 (Im)
  float FC = hi ? 0.f : -c2;   // phase q=2 (Re) / q=1 (Im)
  float FD = hi ? s2 : 0.f;    // phase q=3 (Re) / q=2 (Im)
  // Qubit-0/1 products, indexed by t = kk&3 (compile-time below).
  float AA0 = c0 * c1, AA1 = c0 * s1, AA2 = s0 * c1, AA3 = s0 * s1;

  int row0 = n;        // A-operand row for M-tile 0 (output rows 0..15, Re)
  int row1 = 16 + n;   // A-operand row for M-tile 1 (output rows 16..31, Im)
  const v2f* M2 = (const v2f*)Mg;

  v8f acc0 = {};  // Re psi_out tile
  v8f acc1 = {};  // Im psi_out tile
#pragma unroll
  for (int kk = 0; kk < 8; ++kk) {
    int t = kk & 3;                       // compile-time after unroll
    float At  = (t == 0) ? AA0 : (t == 1) ? AA1 : (t == 2) ? AA2 : AA3;
    int pc0   = (t == 0) ? 0 : (t == 3) ? 2 : 1;   // popcount(t)
    bool isRe = (kk < 4);
    // F(q) table: Re: {FA,FB,FC,FD}[q%4] ; Im: {FB,FC,FD,FA}[q%4]
    int qx = pc0 & 3, qy = (pc0 + 1) & 3;
    float Fx = isRe ? ((qx == 0) ? FA : (qx == 1) ? FB : (qx == 2) ? FC : FD)
                    : ((qx == 0) ? FB : (qx == 1) ? FC : (qx == 2) ? FD : FA);
    float Fy = isRe ? ((qy == 0) ? FA : (qy == 1) ? FB : (qy == 2) ? FC : FD)
                    : ((qy == 0) ? FB : (qy == 1) ? FC : (qy == 2) ? FD : FA);
    v2f bb;
    bb.x = At * c3 * Fx;                  // h = 0
    bb.y = At * s3 * Fy;                  // h = 1

    int kA = 4 * kk + (hi ? 2 : 0);       // A lanes 16-31 carry K=2,3
    v2f a0 = M2[(row0 * 32 + kA) >> 1];
    v2f a1 = M2[(row1 * 32 + kA) >> 1];
    acc0 = __builtin_amdgcn_wmma_f32_16x16x4_f32(false, a0, false, bb,
                                                 (short)0, acc0, false, false);
    acc1 = __builtin_amdgcn_wmma_f32_16x16x4_f32(false, a1, false, bb,
                                                 (short)0, acc1, false, false);
  }

  // lane < 16 holds basis states 0..7 of patch n; lane >= 16 holds 8..15
  float ezv[4];
#pragma unroll
  for (int w = 0; w < 4; ++w) {
    float p = 0.f;
#pragma unroll
    for (int v = 0; v < 8; ++v) {
      float pr = acc0[v] * acc0[v] + acc1[v] * acc1[v];
      float sgn;
      if (w == 0) sgn = hi ? -1.f : 1.f;                       // bit 3 = half
      else        sgn = ((v >> (3 - w)) & 1) ? -1.f : 1.f;     // bits 2..0
      p += sgn * pr;
    }
    ezv[w] = p + __shfl_xor(p, 16, 32);
  }

  if (!hi && j < HP) {
    int base = i * HP + j;
#pragma unroll
    for (int w = 0; w < 4; ++w) ez[w * HP * HP + base] = ezv[w];
  }
}

// ---------------------------------------------------------------------------
// Kernel 2: sigmoid(maxpool2(ez)) -> (4,255,255)
// ---------------------------------------------------------------------------
__global__ void k_pool1(const float* __restrict__ ez, float* __restrict__ out) {
  int idx = blockIdx.x * blockDim.x + threadIdx.x;
  if (idx >= 4 * 255 * 255) return;
  int c = idx / (255 * 255);
  int rem = idx % (255 * 255);
  int oi = rem / 255, oj = rem % 255;
  const float* p = ez + c * HP * HP;
  int i0 = 2 * oi, j0 = 2 * oj;
  float m = p[i0 * HP + j0];
  m = fmaxf(m, p[i0 * HP + j0 + 1]);
  m = fmaxf(m, p[(i0 + 1) * HP + j0]);
  m = fmaxf(m, p[(i0 + 1) * HP + j0 + 1]);
  out[idx] = 1.f / (1.f + __expf(-m));
}

// ---------------------------------------------------------------------------
// Kernel 3: fused conv2(5x5, 4->20) + bias + relu + maxpool2 -> (20,125,125)
// ---------------------------------------------------------------------------
__global__ void k_conv2(const float* __restrict__ in, const float* __restrict__ w,
                        const float* __restrict__ bias, float* __restrict__ out) {
  int idx = blockIdx.x * blockDim.x + threadIdx.x;
  if (idx >= 20 * 125 * 125) return;
  int c = idx / (125 * 125);
  int rem = idx % (125 * 125);
  int oi = rem / 125, oj = rem % 125;
  float a00 = 0.f, a01 = 0.f, a10 = 0.f, a11 = 0.f;
  for (int ic = 0; ic < 4; ++ic) {
    const float* ip = in + ic * 255 * 255 + (2 * oi) * 255 + 2 * oj;
    const float* wp = w + (c * 4 + ic) * 25;
    float win[6][6];
#pragma unroll
    for (int a = 0; a < 6; ++a)
#pragma unroll
      for (int b = 0; b < 6; ++b) win[a][b] = ip[a * 255 + b];
#pragma unroll
    for (int kh = 0; kh < 5; ++kh)
#pragma unroll
      for (int kw = 0; kw < 5; ++kw) {
        float ww = wp[kh * 5 + kw];
        a00 += ww * win[kh][kw];
        a01 += ww * win[kh][kw + 1];
        a10 += ww * win[kh + 1][kw];
        a11 += ww * win[kh + 1][kw + 1];
      }
  }
  float m = fmaxf(fmaxf(a00, a01), fmaxf(a10, a11)) + bias[c];
  out[idx] = fmaxf(m, 0.f);
}

// ---------------------------------------------------------------------------
// Kernel 4: conv3(3x3, 20->40) + bias + relu -> (40,123,123)
// ---------------------------------------------------------------------------
__global__ void k_conv3(const float* __restrict__ in, const float* __restrict__ w,
                        const float* __restrict__ bias, float* __restrict__ out) {
  int idx = blockIdx.x * blockDim.x + threadIdx.x;
  if (idx >= 40 * 123 * 123) return;
  int c = idx / (123 * 123);
  int rem = idx % (123 * 123);
  int oi = rem / 123, oj = rem % 123;
  float acc = bias[c];
  for (int ic = 0; ic < 20; ++ic) {
    const float* ip = in + ic * 125 * 125 + oi * 125 + oj;
    const float* wp = w + (c * 20 + ic) * 9;
    acc += wp[0] * ip[0]       + wp[1] * ip[1]       + wp[2] * ip[2]
         + wp[3] * ip[125]     + wp[4] * ip[126]     + wp[5] * ip[127]
         + wp[6] * ip[250]     + wp[7] * ip[251]     + wp[8] * ip[252];
  }
  out[idx] = fmaxf(acc, 0.f);
}

// ---------------------------------------------------------------------------
// Kernel 5: adaptive max pool 123x123 -> 4x4 per channel (640 cells)
// ---------------------------------------------------------------------------
__global__ void k_amax(const float* __restrict__ in, float* __restrict__ feat) {
  int idx = blockIdx.x * blockDim.x + threadIdx.x;
  if (idx >= 640) return;
  int c = idx >> 4, oi = (idx >> 2) & 3, oj = idx & 3;
  int si = oi * 123 / 4, ei = ((oi + 1) * 123 + 3) / 4;
  int sj = oj * 123 / 4, ej = ((oj + 1) * 123 + 3) / 4;
  const float* p = in + c * 123 * 123;
  float m = -3.402823466e38f;
  for (int a = si; a < ei; ++a)
    for (int b = sj; b < ej; ++b) m = fmaxf(m, p[a * 123 + b]);
  feat[idx] = m;
}

// ---------------------------------------------------------------------------
// Kernel 6: fc1(640->64)+relu then fc2(64->10), single 64-thread block
// ---------------------------------------------------------------------------
__global__ __launch_bounds__(64) void k_fc(const float* __restrict__ feat,
                                           const float* __restrict__ w1,
                                           const float* __restrict__ b1,
                                           const float* __restrict__ w2,
                                           const float* __restrict__ b2,
                                           float* __restrict__ out) {
  __shared__ float f[640];
  __shared__ float h1[64];
  int t = threadIdx.x;
  for (int k = t; k < 640; k += 64) f[k] = feat[k];
  __syncthreads();
  float acc = b1[t];
  for (int k = 0; k < 640; ++k) acc += w1[t * 640 + k] * f[k];
  h1[t] = fmaxf(acc, 0.f);
  __syncthreads();
  if (t < 10) {
    float a2 = b2[t];
    for (int k = 0; k < 64; ++k) a2 += w2[t * 64 + k] * h1[k];
    out[t] = a2;
  }
}

// ---------------------------------------------------------------------------
extern "C" void kernel_launch(void* const* d_in, const int* in_sizes, int n_in,
                              void* d_out, int out_size, void* d_ws, size_t ws_size,
                              hipStream_t stream) {
  const float* x   = (const float*)d_in[0];
  const float* qw  = (const float*)d_in[1];
  const float* w2  = (const float*)d_in[2];
  const float* b2  = (const float*)d_in[3];
  const float* w3  = (const float*)d_in[4];
  const float* b3  = (const float*)d_in[5];
  const float* fw1 = (const float*)d_in[6];
  const float* fb1 = (const float*)d_in[7];
  const float* fw2 = (const float*)d_in[8];
  const float* fb2 = (const float*)d_in[9];
  float* out = (float*)d_out;

  char* ws = (char*)d_ws;
  size_t off = 0;
  auto carve = [&](size_t nfloats) {
    float* p = (float*)(ws + off);
    off = (off + nfloats * sizeof(float) + 255) & ~(size_t)255;
    return p;
  };
  float* Mg   = carve(1024);                 // 32x32 real GEMM matrix
  float* ezb  = carve((size_t)4 * HP * HP);  // raw ez (4,511,511)
  float* p1   = carve((size_t)4 * 255 * 255);
  float* p2   = carve((size_t)20 * 125 * 125);
  float* c3o  = carve((size_t)40 * 123 * 123);
  float* feat = carve(640);

  k_build_M<<<1, 32, 0, stream>>>(qw, Mg);
  // 511 rows * 32 tiles = 16352 waves = 2044 blocks of 8 waves
  k_qconv<<<2044, 256, 0, stream>>>(x, Mg, ezb);
  k_pool1<<<(4 * 255 * 255 + 255) / 256, 256, 0, stream>>>(ezb, p1);
  k_conv2<<<(20 * 125 * 125 + 255) / 256, 256, 0, stream>>>(p1, w2, b2, p2);
  k_conv3<<<(40 * 123 * 123 + 255) / 256, 256, 0, stream>>>(p2, w3, b3, c3o);
  k_amax<<<3, 256, 0, stream>>>(c3o, feat);
  k_fc<<<1, 64, 0, stream>>>(feat, fw1, fb1, fw2, fb2, out);
}